// GNNAnomalyDetector_26018911879667
// MI455X (gfx1250) — compile-verified
//
#include <hip/hip_runtime.h>

#define NODES 100000
#define EDGES 1200000
#define TOTE  (EDGES + NODES)

typedef __attribute__((ext_vector_type(2))) float v2f;
typedef __attribute__((ext_vector_type(8))) float v8f;

// ---------------------------------------------------------------- utilities
__global__ void zero_f32(float* __restrict__ p, int n) {
    int i = blockIdx.x * blockDim.x + threadIdx.x;
    if (i < n) p[i] = 0.0f;
}

__device__ __forceinline__ void edge_ends(const int* __restrict__ src,
                                          const int* __restrict__ dst,
                                          int e, int& s, int& d) {
    if (e < EDGES) { s = src[e]; d = dst[e]; }
    else           { s = d = e - EDGES; }          // self-loop
}

// monotone float<->uint mapping so atomicMax(uint) == float max
__device__ __forceinline__ unsigned fmap(float f) {
    unsigned u = __float_as_uint(f);
    return (u & 0x80000000u) ? ~u : (u | 0x80000000u);
}
__device__ __forceinline__ float funmap(unsigned m) {
    return (m & 0x80000000u) ? __uint_as_float(m ^ 0x80000000u)
                             : __uint_as_float(~m);
}
__device__ __forceinline__ float lrelu(float x) { return x > 0.0f ? x : 0.2f * x; }

// ---------------------------------------------------------------- degree
__global__ void deg_kernel(const int* __restrict__ src, const int* __restrict__ dst,
                           float* __restrict__ deg) {
    int e = blockIdx.x * blockDim.x + threadIdx.x;
    if (e >= TOTE) return;
    int d = (e < EDGES) ? dst[e] : (e - EDGES);
    atomicAdd(&deg[d], 1.0f);
}
__global__ void dinv_kernel(float* __restrict__ deg) {
    int i = blockIdx.x * blockDim.x + threadIdx.x;
    if (i < NODES) deg[i] = rsqrtf(deg[i]);       // deg >= 1 (self-loop)
}

// ---------------------------------------------------------------- WMMA GEMM
// C[M x Cdim] = A[M x K] @ W[K x Cdim], row-major, M % 16 == 0, K,Cdim % 16 == 0.
// One wave32 computes a 16x16 output tile via K/4 V_WMMA_F32_16X16X4_F32 steps.
__global__ __launch_bounds__(256)
void gemm_wmma(const float* __restrict__ A, const float* __restrict__ W,
               float* __restrict__ Cout, int K, int Cdim, int totalWaves) {
    int wave = blockIdx.x * (blockDim.x >> 5) + (threadIdx.x >> 5);
    if (wave >= totalWaves) return;               // wave-uniform: EXEC stays full
    int tilesN = Cdim >> 4;
    int tm = wave / tilesN, tn = wave % tilesN;
    int lane = threadIdx.x & 31;
    int half = lane >> 4;                          // lane group 0..15 / 16..31
    int lr   = lane & 15;
    int arow = tm * 16 + lr;                       // A: M = lr (both halves)
    int bcol = tn * 16 + lr;                       // B: N = lr (both halves)

    v8f acc = {};
    const float* ap = A + (long)arow * K + 2 * half;          // K=2*half, 2*half+1
    const float* wp = W + (long)(2 * half) * Cdim + bcol;
    for (int k = 0; k < K; k += 4) {
        v2f a; a[0] = ap[0];    a[1] = ap[1];
        v2f b; b[0] = wp[0];    b[1] = wp[Cdim];
        acc = __builtin_amdgcn_wmma_f32_16x16x4_f32(
            /*neg_a=*/false, a, /*neg_b=*/false, b,
            /*c_mod=*/(short)0, acc, /*reuse_a=*/false, /*reuse_b=*/false);
        ap += 4;
        wp += 4 * (long)Cdim;
    }
#pragma unroll
    for (int r = 0; r < 8; r++)                    // D: VGPR r = row r (+8 for half 1)
        Cout[(long)(tm * 16 + r + 8 * half) * Cdim + bcol] = acc[r];
}

// ---------------------------------------------------------------- GCN scatter
// out[dst] += h[src] * dinv[src]*dinv[dst]; 16 consecutive lanes cover one
// edge's channels in float4 chunks -> coalesced gather + f32 atomic scatter.
__global__ void gcn_agg(const float* __restrict__ h, const int* __restrict__ src,
                        const int* __restrict__ dst, const float* __restrict__ dinv,
                        float* __restrict__ out, int Cdim) {
    int groups = Cdim >> 2;
    long tid = (long)blockIdx.x * blockDim.x + threadIdx.x;
    if (tid >= (long)TOTE * groups) return;
    int e  = (int)(tid / groups);
    int c4 = (int)(tid % groups) << 2;
    int s, d; edge_ends(src, dst, e, s, d);
    float nrm = dinv[s] * dinv[d];
    float4 hv = *(const float4*)(h + (long)s * Cdim + c4);
    float* op = out + (long)d * Cdim + c4;
    atomicAdd(op + 0, hv.x * nrm);
    atomicAdd(op + 1, hv.y * nrm);
    atomicAdd(op + 2, hv.z * nrm);
    atomicAdd(op + 3, hv.w * nrm);
}

__global__ void bias_relu(float* __restrict__ h, const float* __restrict__ b,
                          int cmask, long n) {
    long i = (long)blockIdx.x * blockDim.x + threadIdx.x;
    if (i >= n) return;
    float v = h[i] + b[i & cmask];
    h[i] = v > 0.0f ? v : 0.0f;
}

// ---------------------------------------------------------------- GAT
__global__ void gat_scores(const float* __restrict__ g, const float* __restrict__ att_s,
                           const float* __restrict__ att_d, float* __restrict__ as_,
                           float* __restrict__ ad_) {
    int t = blockIdx.x * blockDim.x + threadIdx.x;
    if (t >= NODES * 4) return;
    int n = t >> 2, h = t & 3;
    const float* gp = g + (long)n * 64 + h * 16;
    float s = 0.0f, d = 0.0f;
#pragma unroll
    for (int c = 0; c < 16; c++) {
        float v = gp[c];
        s += v * att_s[h * 16 + c];
        d += v * att_d[h * 16 + c];
    }
    as_[t] = s; ad_[t] = d;
}

__global__ void attn_max(const int* __restrict__ src, const int* __restrict__ dst,
                         const float* __restrict__ as_, const float* __restrict__ ad_,
                         unsigned* __restrict__ m) {
    long t = (long)blockIdx.x * blockDim.x + threadIdx.x;
    if (t >= (long)TOTE * 4) return;
    int e = (int)(t >> 2), h = (int)(t & 3);
    int s, d; edge_ends(src, dst, e, s, d);
    float ee = lrelu(as_[s * 4 + h] + ad_[d * 4 + h]);
    atomicMax(&m[d * 4 + h], fmap(ee));
}

__global__ void attn_den(const int* __restrict__ src, const int* __restrict__ dst,
                         const float* __restrict__ as_, const float* __restrict__ ad_,
                         const unsigned* __restrict__ m, float* __restrict__ den) {
    long t = (long)blockIdx.x * blockDim.x + threadIdx.x;
    if (t >= (long)TOTE * 4) return;
    int e = (int)(t >> 2), h = (int)(t & 3);
    int s, d; edge_ends(src, dst, e, s, d);
    float ee = lrelu(as_[s * 4 + h] + ad_[d * 4 + h]);
    atomicAdd(&den[d * 4 + h], expf(ee - funmap(m[d * 4 + h])));
}

__global__ void attn_agg(const int* __restrict__ src, const int* __restrict__ dst,
                         const float* __restrict__ as_, const float* __restrict__ ad_,
                         const unsigned* __restrict__ m, const float* __restrict__ den,
                         const float* __restrict__ g, float* __restrict__ o) {
    long t = (long)blockIdx.x * blockDim.x + threadIdx.x;
    if (t >= (long)TOTE * 4) return;
    int e = (int)(t >> 2), h = (int)(t & 3);
    int s, d; edge_ends(src, dst, e, s, d);
    float ee = lrelu(as_[s * 4 + h] + ad_[d * 4 + h]);
    float alpha = expf(ee - funmap(m[d * 4 + h])) / den[d * 4 + h];
    const float* gp = g + (long)s * 64 + h * 16;
    float* op = o + (long)d * 64 + h * 16;
#pragma unroll
    for (int c = 0; c < 16; c += 4) {
        float4 gv = *(const float4*)(gp + c);
        atomicAdd(op + c + 0, gv.x * alpha);
        atomicAdd(op + c + 1, gv.y * alpha);
        atomicAdd(op + c + 2, gv.z * alpha);
        atomicAdd(op + c + 3, gv.w * alpha);
    }
}

// ---------------------------------------------------------------- pool + MLP
__global__ void mean_pool(const float* __restrict__ o, float* __restrict__ pooled) {
    __shared__ float sd[256];
    int c = blockIdx.x;                            // 0..63
    float s = 0.0f;
    for (int n = threadIdx.x; n < NODES; n += blockDim.x)
        s += o[(long)n * 64 + c];
    sd[threadIdx.x] = s;
    __syncthreads();
    for (int st = 128; st > 0; st >>= 1) {
        if ((int)threadIdx.x < st) sd[threadIdx.x] += sd[threadIdx.x + st];
        __syncthreads();
    }
    if (threadIdx.x == 0) pooled[c] = sd[0] / (float)NODES;
}

__global__ void classifier(const float* __restrict__ pooled, const float* __restrict__ Wc1,
                           const float* __restrict__ bc1, const float* __restrict__ Wc2,
                           const float* __restrict__ bc2, float* __restrict__ out) {
    __shared__ float z[32];
    int j = threadIdx.x;
    if (j < 32) {
        float s = bc1[j];
        for (int i = 0; i < 64; i++) s += pooled[i] * Wc1[i * 32 + j];
        z[j] = s > 0.0f ? s : 0.0f;
    }
    __syncthreads();
    if (j < 2) {
        float s = bc2[j];
        for (int q = 0; q < 32; q++) s += z[q] * Wc2[q * 2 + j];
        out[j] = s;
    }
}

// ---------------------------------------------------------------- launch
static inline unsigned gblocks(long n) { return (unsigned)((n + 255) / 256); }

extern "C" void kernel_launch(void* const* d_in, const int* in_sizes, int n_in,
                              void* d_out, int out_size, void* d_ws, size_t ws_size,
                              hipStream_t stream) {
    const float* x    = (const float*)d_in[0];
    const int*   ei   = (const int*)d_in[1];
    const int*   src  = ei;                // edge_index[0]
    const int*   dst  = ei + EDGES;        // edge_index[1]
    const float* W1   = (const float*)d_in[2];
    const float* b1   = (const float*)d_in[3];
    const float* W2   = (const float*)d_in[4];
    const float* b2   = (const float*)d_in[5];
    const float* W3   = (const float*)d_in[6];
    const float* b3   = (const float*)d_in[7];
    const float* Wg   = (const float*)d_in[8];
    const float* atts = (const float*)d_in[9];
    const float* attd = (const float*)d_in[10];
    const float* bg   = (const float*)d_in[11];
    const float* Wc1  = (const float*)d_in[12];
    const float* bc1  = (const float*)d_in[13];
    const float* Wc2  = (const float*)d_in[14];
    const float* bc2  = (const float*)d_in[15];

    // workspace carve-up (~58 MB)
    float*    P    = (float*)d_ws;                 // N*64  (GEMM outputs)
    float*    Q    = P + (long)NODES * 64;         // N*64  (agg outputs / GEMM inputs / GAT out)
    float*    Dv   = Q + (long)NODES * 64;         // N     (deg -> deg^-1/2)
    float*    AS   = Dv + NODES;                   // N*4
    float*    AD   = AS + (long)NODES * 4;         // N*4
    unsigned* Mx   = (unsigned*)(AD + (long)NODES * 4);   // N*4 (mapped max)
    float*    DEN  = (float*)(Mx + (long)NODES * 4);      // N*4
    float*    POOL = DEN + (long)NODES * 4;        // 64

    dim3 B(256);

    // degrees -> deg^-1/2
    zero_f32<<<gblocks(NODES), B, 0, stream>>>(Dv, NODES);
    deg_kernel<<<gblocks(TOTE), B, 0, stream>>>(src, dst, Dv);
    dinv_kernel<<<gblocks(NODES), B, 0, stream>>>(Dv);

    const int tw64 = (NODES / 16) * (64 / 16);     // waves for Cout=64
    const int tw32 = (NODES / 16) * (32 / 16);     // waves for Cout=32

    // GCN layer 1: P = x@W1 ; Q = agg(P) ; Q = relu(Q+b1)
    gemm_wmma<<<dim3((tw64 + 7) / 8), B, 0, stream>>>(x, W1, P, 64, 64, tw64);
    zero_f32<<<gblocks((long)NODES * 64), B, 0, stream>>>(Q, NODES * 64);
    gcn_agg<<<gblocks((long)TOTE * 16), B, 0, stream>>>(P, src, dst, Dv, Q, 64);
    bias_relu<<<gblocks((long)NODES * 64), B, 0, stream>>>(Q, b1, 63, (long)NODES * 64);

    // GCN layer 2
    gemm_wmma<<<dim3((tw64 + 7) / 8), B, 0, stream>>>(Q, W2, P, 64, 64, tw64);
    zero_f32<<<gblocks((long)NODES * 64), B, 0, stream>>>(Q, NODES * 64);
    gcn_agg<<<gblocks((long)TOTE * 16), B, 0, stream>>>(P, src, dst, Dv, Q, 64);
    bias_relu<<<gblocks((long)NODES * 64), B, 0, stream>>>(Q, b2, 63, (long)NODES * 64);

    // GCN layer 3 (64 -> 32)
    gemm_wmma<<<dim3((tw32 + 7) / 8), B, 0, stream>>>(Q, W3, P, 64, 32, tw32);
    zero_f32<<<gblocks((long)NODES * 32), B, 0, stream>>>(Q, NODES * 32);
    gcn_agg<<<gblocks((long)TOTE * 8), B, 0, stream>>>(P, src, dst, Dv, Q, 32);
    bias_relu<<<gblocks((long)NODES * 32), B, 0, stream>>>(Q, b3, 31, (long)NODES * 32);

    // GAT: g = Q@Wg (N x 64) into P; per-node head scores
    gemm_wmma<<<dim3((tw64 + 7) / 8), B, 0, stream>>>(Q, Wg, P, 32, 64, tw64);
    gat_scores<<<gblocks((long)NODES * 4), B, 0, stream>>>(P, atts, attd, AS, AD);

    // segment softmax + weighted aggregation into Q (h3 no longer needed)
    zero_f32<<<gblocks((long)NODES * 4), B, 0, stream>>>((float*)Mx, NODES * 4); // bits 0 < fmap(any)
    zero_f32<<<gblocks((long)NODES * 4), B, 0, stream>>>(DEN, NODES * 4);
    zero_f32<<<gblocks((long)NODES * 64), B, 0, stream>>>(Q, NODES * 64);
    attn_max<<<gblocks((long)TOTE * 4), B, 0, stream>>>(src, dst, AS, AD, Mx);
    attn_den<<<gblocks((long)TOTE * 4), B, 0, stream>>>(src, dst, AS, AD, Mx, DEN);
    attn_agg<<<gblocks((long)TOTE * 4), B, 0, stream>>>(src, dst, AS, AD, Mx, DEN, P, Q);
    bias_relu<<<gblocks((long)NODES * 64), B, 0, stream>>>(Q, bg, 63, (long)NODES * 64);

    // mean pool + classifier -> d_out[2]
    mean_pool<<<dim3(64), B, 0, stream>>>(Q, POOL);
    classifier<<<dim3(1), dim3(64), 0, stream>>>(POOL, Wc1, bc1, Wc2, bc2, (float*)d_out);
}